// BERTVideoModel_68934225101528
// MI455X (gfx1250) — compile-verified
//
#include <hip/hip_runtime.h>
#include <hip/hip_bf16.h>

// ---------------- model constants ----------------
#define DIM   1024
#define NHEAD 16
#define HDIM  64
#define BF    16        // frames
#define PP    1024      // patches per frame
#define NTOK  (1 + BF * PP)   // 16385
#define LNUM  4
#define KPAT  768       // S*S*C
#define EPSLN 1e-5f

typedef float v2f __attribute__((ext_vector_type(2)));
typedef float v8f __attribute__((ext_vector_type(8)));

// =====================================================================
// WMMA fp32 GEMM:  Out[m][n] = sum_k A[m][k] * B(k,n)  (+bias[n]) (+Add[m][n])
//   BT = false : B is row-major [K x 1024]            (X @ W)
//   BT = true  : B is row-major [1024 x K], use B[n][k] (X @ W^T)
// Block: 256 thr = 8 waves, tile 128(M) x 64(N), K staged 16 at a time.
// Wave w computes rows [w*16, w*16+16) x all 64 cols = 4 accum tiles.
// Fragment layouts per CDNA5 ISA (32-bit A 16x4 / B 4x16 / C 16x16):
//   lanes 0-15 : K = {k, k+1}; lanes 16-31 : K = {k+2, k+3}
//   C/D: vgpr r -> row r (lanes 0-15) / row r+8 (lanes 16-31), col = lane%16
// B is staged PAIRED in LDS: Bs[kp][col][2] with kp=k/2 so each fragment is
// one contiguous 8B pair -> single ds_load_b64, no v_mov repacking.
// =====================================================================
#define BSTR 98   // pair-row stride in floats (even; 98 mod 64 = 34 -> bank spread)

template<bool BT, bool HAS_BIAS, bool HAS_ADD>
__global__ __launch_bounds__(256)
void gemm_wmma(const float* __restrict__ A, const float* __restrict__ Bm,
               const float* __restrict__ bias, const float* __restrict__ Add,
               float* __restrict__ Out, int M, int K)
{
    __shared__ __align__(16) float As[128 * 20];  // stride 20 floats (80B, 16B-aligned)
    __shared__ __align__(16) float Bs[8 * BSTR];  // paired layout [kp][col][2]

    const int tid  = threadIdx.x;
    const int wave = tid >> 5;
    const int lane = tid & 31;
    const int l16  = lane & 15;
    const int lhf  = lane >> 4;          // 0 or 1
    const int tm   = blockIdx.x * 128;
    const int tn   = blockIdx.y * 64;

    v8f acc[4];
    for (int t = 0; t < 4; ++t)
        for (int i = 0; i < 8; ++i) acc[t][i] = 0.0f;

    for (int k0 = 0; k0 < K; k0 += 16) {
        // ---- stage A tile 128x16 ----
        {
            const int r  = tid >> 2;
            const int c4 = (tid & 3) * 4;
            for (int i = 0; i < 2; ++i) {
                const int rr = r + i * 64;
                int gr = tm + rr; if (gr >= M) gr = M - 1;   // clamp tail
                const float4 v = *(const float4*)(A + (size_t)gr * K + k0 + c4);
                *(float4*)(&As[rr * 20 + c4]) = v;
            }
            // software prefetch of next A tile (global_prefetch_b8)
            if (k0 + 16 < K) {
                int gr = tm + r; if (gr >= M) gr = M - 1;
                __builtin_prefetch(A + (size_t)gr * K + k0 + 16 + c4, 0, 0);
            }
        }
        // ---- stage B tile 16x64 into paired layout ----
        if (BT) {
            // Bm is [1024 x K]; Bs[kp][j] pair = (Bm[tn+j][k], Bm[tn+j][k+1])
            const int j  = tid & 63;
            const int i4 = (tid >> 6) * 4;   // k offset: 0,4,8,12
            const float4 v = *(const float4*)(Bm + (size_t)(tn + j) * K + k0 + i4);
            v2f p0; p0.x = v.x; p0.y = v.y;
            v2f p1; p1.x = v.z; p1.y = v.w;
            *(v2f*)(&Bs[((i4 >> 1) + 0) * BSTR + 2 * j]) = p0;
            *(v2f*)(&Bs[((i4 >> 1) + 1) * BSTR + 2 * j]) = p1;
        } else {
            // Bm is [K x 1024]; thread owns (kp, two cols)
            const int kp = tid >> 5;          // 0..7
            const int j2 = (tid & 31) * 2;    // col pair base 0..62
            const float2 r0 = *(const float2*)(Bm + (size_t)(k0 + 2 * kp)     * DIM + tn + j2);
            const float2 r1 = *(const float2*)(Bm + (size_t)(k0 + 2 * kp + 1) * DIM + tn + j2);
            v2f p0; p0.x = r0.x; p0.y = r1.x;
            v2f p1; p1.x = r0.y; p1.y = r1.y;
            *(v2f*)(&Bs[kp * BSTR + 2 * j2])     = p0;
            *(v2f*)(&Bs[kp * BSTR + 2 * j2 + 2]) = p1;
        }
        __syncthreads();

        const int arow = wave * 16 + l16;
        for (int kk = 0; kk < 16; kk += 4) {
            const v2f a = *(const v2f*)(&As[arow * 20 + kk + 2 * lhf]);       // ds_load_b64
            const int kprow = ((kk >> 1) + lhf) * BSTR;
            for (int t = 0; t < 4; ++t) {
                const v2f b = *(const v2f*)(&Bs[kprow + 2 * (t * 16 + l16)]); // ds_load_b64
                acc[t] = __builtin_amdgcn_wmma_f32_16x16x4_f32(
                    false, a, false, b, (short)0, acc[t], false, false);
            }
        }
        __syncthreads();
    }

    // ---- epilogue ----
    for (int t = 0; t < 4; ++t) {
        const int gc = tn + t * 16 + l16;
        for (int rr = 0; rr < 8; ++rr) {
            const int gr = tm + wave * 16 + rr + 8 * lhf;
            if (gr < M) {
                float v = acc[t][rr];
                if (HAS_BIAS) v += bias[gc];
                if (HAS_ADD)  v += Add[(size_t)gr * DIM + gc];
                Out[(size_t)gr * DIM + gc] = v;
            }
        }
    }
}

// =====================================================================
// LayerNorm (optionally preceded by LeakyReLU(0.01), optionally emitting
// per-head sum-of-squares of the normalized output for self-dot scores).
// One block per token row, 256 thr x float4.
// =====================================================================
template<bool LEAKY, bool WRITE_SS>
__global__ __launch_bounds__(256)
void layernorm_k(const float* __restrict__ X, float* __restrict__ Y,
                 const float* __restrict__ g, const float* __restrict__ b,
                 float* __restrict__ ss)
{
    __shared__ float red[256];
    __shared__ float s_mu, s_rstd;
    const int row = blockIdx.x;
    const int tid = threadIdx.x;

    float4 v = *(const float4*)(X + (size_t)row * DIM + tid * 4);
    if (LEAKY) {
        v.x = v.x > 0.f ? v.x : 0.01f * v.x;
        v.y = v.y > 0.f ? v.y : 0.01f * v.y;
        v.z = v.z > 0.f ? v.z : 0.01f * v.z;
        v.w = v.w > 0.f ? v.w : 0.01f * v.w;
    }
    red[tid] = v.x + v.y + v.z + v.w;
    __syncthreads();
    for (int o = 128; o > 0; o >>= 1) { if (tid < o) red[tid] += red[tid + o]; __syncthreads(); }
    if (tid == 0) s_mu = red[0] * (1.0f / DIM);
    __syncthreads();
    const float mu = s_mu;
    const float dx = v.x - mu, dy = v.y - mu, dz = v.z - mu, dw = v.w - mu;
    red[tid] = dx * dx + dy * dy + dz * dz + dw * dw;
    __syncthreads();
    for (int o = 128; o > 0; o >>= 1) { if (tid < o) red[tid] += red[tid + o]; __syncthreads(); }
    if (tid == 0) s_rstd = rsqrtf(red[0] * (1.0f / DIM) + EPSLN);
    __syncthreads();
    const float r = s_rstd;
    const float4 gg = *(const float4*)(g + tid * 4);
    const float4 bb = *(const float4*)(b + tid * 4);
    float4 y;
    y.x = dx * r * gg.x + bb.x;
    y.y = dy * r * gg.y + bb.y;
    y.z = dz * r * gg.z + bb.z;
    y.w = dw * r * gg.w + bb.w;
    *(float4*)(Y + (size_t)row * DIM + tid * 4) = y;

    if (WRITE_SS) {
        __syncthreads();
        red[tid] = y.x * y.x + y.y * y.y + y.z * y.z + y.w * y.w;
        __syncthreads();
        if ((tid & 15) == 0) {
            float hs = 0.f;
            for (int i = 0; i < 16; ++i) hs += red[tid + i];
            ss[(size_t)row * NHEAD + (tid >> 4)] = hs;   // 16 thr -> 1 head (64 elems)
        }
    }
}

// sums of 6 weight tensors (NH*HD*HD each) -> 6 scalars
__global__ __launch_bounds__(256)
void sum6_k(const float* p0, const float* p1, const float* p2,
            const float* p3, const float* p4, const float* p5, float* out)
{
    const float* ps[6] = { p0, p1, p2, p3, p4, p5 };
    __shared__ float red[256];
    const float* p = ps[blockIdx.x];
    float s = 0.f;
    for (int i = threadIdx.x; i < NHEAD * HDIM * HDIM; i += 256) s += p[i];
    red[threadIdx.x] = s;
    __syncthreads();
    for (int o = 128; o > 0; o >>= 1) { if (threadIdx.x < o) red[threadIdx.x] += red[threadIdx.x + o]; __syncthreads(); }
    if (threadIdx.x == 0) out[blockIdx.x] = red[0];
}

// softmax over (nkeys+1) self-dot logits per (group, head)
// token(g, j) = (j==0) ? 0 : 1 + g*gstride + (j-1)
__global__ __launch_bounds__(128)
void attn_softmax_k(const float* __restrict__ ss, const float* __restrict__ scal,
                    int qi, int ki, float* __restrict__ att, int gstride, int nkeys)
{
    __shared__ float red[128];
    const int g = blockIdx.x, h = blockIdx.y, tid = threadIdx.x;
    const int nj = nkeys + 1;
    const float scale = scal[qi] * scal[ki] * 0.125f;  // / sqrt(64)
    float* arow = att + ((size_t)g * NHEAD + h) * nj;

    float mx = -3.4e38f;
    for (int j = tid; j < nj; j += 128) {
        const int t = (j == 0) ? 0 : (1 + g * gstride + (j - 1));
        mx = fmaxf(mx, scale * ss[(size_t)t * NHEAD + h]);
    }
    red[tid] = mx; __syncthreads();
    for (int o = 64; o > 0; o >>= 1) { if (tid < o) red[tid] = fmaxf(red[tid], red[tid + o]); __syncthreads(); }
    mx = red[0]; __syncthreads();

    float sum = 0.f;
    for (int j = tid; j < nj; j += 128) {
        const int t = (j == 0) ? 0 : (1 + g * gstride + (j - 1));
        const float e = __expf(scale * ss[(size_t)t * NHEAD + h] - mx);
        arow[j] = e;
        sum += e;
    }
    red[tid] = sum; __syncthreads();
    for (int o = 64; o > 0; o >>= 1) { if (tid < o) red[tid] += red[tid + o]; __syncthreads(); }
    const float inv = 1.0f / red[0];
    for (int j = tid; j < nj; j += 128) arow[j] *= inv;
}

// resp rows: OM[1+r][h*64+d] = sv*( att[g][h][1+key]*ln[1+r][..] + att[g][h][0]*ln[0][..] )
__global__ __launch_bounds__(256)
void resp_k(const float* __restrict__ ln, const float* __restrict__ att,
            const float* __restrict__ scal, int svi,
            float* __restrict__ OM, int gstride, int nkeys)
{
    const int r = blockIdx.x;          // 0..16383
    const int tid = threadIdx.x;       // col/4
    const int h = tid >> 4;
    const int nj = nkeys + 1;
    const int g = r / gstride, key = r % gstride;
    const float sv = scal[svi];
    const float* ar = att + ((size_t)g * NHEAD + h) * nj;
    const float a0 = ar[0], a1 = ar[1 + key];
    const float4 x  = *(const float4*)(ln + (size_t)(1 + r) * DIM + tid * 4);
    const float4 x0 = *(const float4*)(ln + tid * 4);
    float4 y;
    y.x = sv * (a1 * x.x + a0 * x0.x);
    y.y = sv * (a1 * x.y + a0 * x0.y);
    y.z = sv * (a1 * x.z + a0 * x0.z);
    y.w = sv * (a1 * x.w + a0 * x0.w);
    *(float4*)(OM + (size_t)(1 + r) * DIM + tid * 4) = y;
}

// tok row 0, stage 1: partial column sums over 256-token chunks (deterministic)
__global__ __launch_bounds__(256)
void tok_part_k(const float* __restrict__ ln, const float* __restrict__ att,
                float* __restrict__ part, int gstride, int nkeys)
{
    const int col = blockIdx.x * 256 + threadIdx.x;   // grid.x = 4
    const int chunk = blockIdx.y;                     // 64 chunks
    const int h = col >> 6;
    const int nj = nkeys + 1;
    const int r0 = chunk * 256;
    float acc = 0.f;
    for (int i = 0; i < 256; ++i) {
        const int r = r0 + i;
        const float a = att[((size_t)(r / gstride) * NHEAD + h) * nj];  // att[g][h][0]
        acc += a * ln[(size_t)(1 + r) * DIM + col];
    }
    part[(size_t)chunk * DIM + col] = acc;
}

// tok stage 2: OM[0][col] = sv * (ln[0][col] + sum_chunks part)
__global__ __launch_bounds__(256)
void tok_reduce_k(const float* __restrict__ ln, const float* __restrict__ part,
                  const float* __restrict__ scal, int svi, float* __restrict__ OM)
{
    const int col = blockIdx.x * 256 + threadIdx.x;
    float s = ln[col];
    for (int c = 0; c < 64; ++c) s += part[(size_t)c * DIM + col];
    OM[col] = scal[svi] * s;
}

// patchify: x[B,C,512,512] -> patches[16384][768] (e = c*256 + sy*16 + sx)
__global__ __launch_bounds__(256)
void patchify_k(const float* __restrict__ x, float* __restrict__ patches)
{
    const int m  = blockIdx.x;              // b*1024 + hp*32 + wp
    const int b  = m >> 10;
    const int hp = (m >> 5) & 31;
    const int wp = m & 31;
    const int tid = threadIdx.x;            // sy*16 + sx
    const int row  = hp * 16 + (tid >> 4);
    const int colx = wp * 16 + (tid & 15);
    for (int c = 0; c < 3; ++c)
        patches[(size_t)m * KPAT + c * 256 + tid] =
            x[(((size_t)b * 3 + c) * 512 + row) * 512 + colx];
}

__global__ __launch_bounds__(256)
void cls_k(const float* __restrict__ cls, float* __restrict__ emb)
{
    const int i = blockIdx.x * 256 + threadIdx.x;
    if (i < DIM) emb[i] = cls[i];
}

// =====================================================================
extern "C" void kernel_launch(void* const* d_in, const int* in_sizes, int n_in,
                              void* d_out, int out_size, void* d_ws, size_t ws_size,
                              hipStream_t stream)
{
    (void)in_sizes; (void)n_in; (void)out_size; (void)ws_size;
    const float* x       = (const float*)d_in[0];
    const float* cls     = (const float*)d_in[1];
    const float* w_embed = (const float*)d_in[2];
    const float* pos_enc = (const float*)d_in[3];
    const float* ln_t_g  = (const float*)d_in[4];
    const float* ln_t_b  = (const float*)d_in[5];
    const float* wq_t    = (const float*)d_in[6];
    const float* wk_t    = (const float*)d_in[7];
    const float* wv_t    = (const float*)d_in[8];
    const float* wt_t    = (const float*)d_in[9];
    const float* ln_s_g  = (const float*)d_in[10];
    const float* ln_s_b  = (const float*)d_in[11];
    const float* wq_s    = (const float*)d_in[12];
    const float* wk_s    = (const float*)d_in[13];
    const float* wv_s    = (const float*)d_in[14];
    const float* wt_s    = (const float*)d_in[15];
    const float* ln_m_g  = (const float*)d_in[16];
    const float* ln_m_b  = (const float*)d_in[17];
    const float* mlp_w   = (const float*)d_in[18];
    const float* mlp_b   = (const float*)d_in[19];
    const float* out_g   = (const float*)d_in[20];
    const float* out_b   = (const float*)d_in[21];

    // workspace layout (floats): ~271 MB total
    float* ws  = (float*)d_ws;
    const size_t ND = (size_t)NTOK * DIM;             // 16,778,240
    float* E0   = ws;
    float* E1   = E0 + ND;
    float* LNB  = E1 + ND;
    float* OM   = LNB + ND;                            // also holds patches at start
    float* SS   = OM + ND;                             // NTOK * NH
    float* ATT  = SS + (size_t)NTOK * NHEAD;           // max 1024*16*17
    float* PART = ATT + (size_t)PP * NHEAD * 17;       // 64 * DIM
    float* SCAL = PART + 64 * DIM;                     // 6 scalars

    const size_t WSZ = (size_t)NHEAD * HDIM * HDIM;    // 65536 per q/k/v slice
    const size_t DSQ = (size_t)DIM * DIM;

    // ---- patch embedding ----
    patchify_k<<<BF * PP, 256, 0, stream>>>(x, OM);
    cls_k<<<4, 256, 0, stream>>>(cls, E0);
    {
        dim3 ge(128, 16);   // M = 16384 exact
        gemm_wmma<true, false, true><<<ge, 256, 0, stream>>>(
            OM, w_embed, nullptr, pos_enc, E0 + DIM, BF * PP, KPAT);
    }

    float* cur = E0;
    float* alt = E1;
    const dim3 gg(129, 16);     // ceil(16385/128) x 1024/64

    for (int l = 0; l < LNUM; ++l) {
        sum6_k<<<6, 256, 0, stream>>>(wq_t + l * WSZ, wk_t + l * WSZ, wv_t + l * WSZ,
                                      wq_s + l * WSZ, wk_s + l * WSZ, wv_s + l * WSZ, SCAL);
        // ---- temporal ----
        layernorm_k<false, true><<<NTOK, 256, 0, stream>>>(cur, LNB, ln_t_g + l * DIM, ln_t_b + l * DIM, SS);
        attn_softmax_k<<<dim3(PP, NHEAD), 128, 0, stream>>>(SS, SCAL, 0, 1, ATT, 16, 16);
        resp_k<<<BF * PP, 256, 0, stream>>>(LNB, ATT, SCAL, 2, OM, 16, 16);
        tok_part_k<<<dim3(4, 64), 256, 0, stream>>>(LNB, ATT, PART, 16, 16);
        tok_reduce_k<<<4, 256, 0, stream>>>(LNB, PART, SCAL, 2, OM);
        gemm_wmma<false, false, true><<<gg, 256, 0, stream>>>(
            OM, wt_t + (size_t)l * DSQ, nullptr, cur, alt, NTOK, DIM);
        { float* t = cur; cur = alt; alt = t; }
        // ---- spatial ----
        layernorm_k<false, true><<<NTOK, 256, 0, stream>>>(cur, LNB, ln_s_g + l * DIM, ln_s_b + l * DIM, SS);
        attn_softmax_k<<<dim3(BF, NHEAD), 128, 0, stream>>>(SS, SCAL, 3, 4, ATT, 1024, 1024);
        resp_k<<<BF * PP, 256, 0, stream>>>(LNB, ATT, SCAL, 5, OM, 1024, 1024);
        tok_part_k<<<dim3(4, 64), 256, 0, stream>>>(LNB, ATT, PART, 1024, 1024);
        tok_reduce_k<<<4, 256, 0, stream>>>(LNB, PART, SCAL, 5, OM);
        gemm_wmma<false, false, true><<<gg, 256, 0, stream>>>(
            OM, wt_s + (size_t)l * DSQ, nullptr, cur, alt, NTOK, DIM);
        { float* t = cur; cur = alt; alt = t; }
        // ---- MLP + LeakyReLU + out-LN ----
        layernorm_k<false, false><<<NTOK, 256, 0, stream>>>(cur, LNB, ln_m_g + l * DIM, ln_m_b + l * DIM, nullptr);
        gemm_wmma<true, true, true><<<gg, 256, 0, stream>>>(
            LNB, mlp_w + (size_t)l * DSQ, mlp_b + l * DIM, cur, alt, NTOK, DIM);
        float* dest = (l == LNUM - 1) ? (float*)d_out : cur;
        layernorm_k<true, false><<<NTOK, 256, 0, stream>>>(alt, dest, out_g + l * DIM, out_b + l * DIM, nullptr);
        // cur now holds the new embedding (unless final layer wrote d_out)
    }
}